// GraphPyramidPooling_38843684225051
// MI455X (gfx1250) — compile-verified
//
#include <hip/hip_runtime.h>
#include <math.h>

// MI455X / gfx1250 implementation of GraphPyramidPooling.
//
// Key insight: the adjacency g never influences the returned hs (it only
// produces cg, which is dead w.r.t. the output), so the N^3 boolean matmul
// chain is skipped entirely. Remaining work is 3 GEMVs + exact top-k ranking
// + gather/gate/scatter: ~45 MB of traffic -> ~2us at 23.3 TB/s (bandwidth
// bound). GEMVs use V_WMMA_F32_16X16X4_F32 to keep full fp32 precision
// (top-k ordering is sensitive to score error) while using the matrix pipe.

typedef float v2f __attribute__((ext_vector_type(2)));
typedef float v8f __attribute__((ext_vector_type(8)));

#define N_FULL 4096
#define DIM    512

// scores[r] = sigmoid(gate[r] * dot(h[sel[r]], w) + bias), one wave per 16 rows.
// A (16x4 f32): lanes 0-15 hold M=lane, {VGPR0,VGPR1}={K0,K1}; lanes 16-31 {K2,K3}.
// B (4x16 f32): w[k] broadcast along N; V0/V1 lanes0-15 = K0/K1, lanes16-31 = K2/K3.
// D column 0: lane 0 has rows base+0..7 in c[0..7], lane 16 has rows base+8..15.
__global__ __launch_bounds__(32)
void score_wmma_kernel(const float* __restrict__ h,
                       const int*   __restrict__ sel,    // null at level 0
                       const float* __restrict__ gate,   // null at level 0
                       const float* __restrict__ w,
                       const float* __restrict__ bvec, int level,
                       float* __restrict__ scores, int n)
{
    const int lane = threadIdx.x;            // 0..31
    const int base = blockIdx.x * 16;
    if (base >= n) return;                   // uniform per wave: EXEC stays all-1s
    const int hi   = lane >> 4;              // 0 or 1
    const int m    = lane & 15;
    int row  = base + m;
    int lrow = row < n ? row : n - 1;        // clamp loads for tail rows
    int srow = sel ? sel[lrow] : lrow;

    const float* hrow = h + (size_t)srow * DIM + 2 * hi;
    const float* wptr = w + 2 * hi;

    v8f c = {};
#pragma unroll 4
    for (int k = 0; k < DIM; k += 4) {
        v2f a = *(const v2f*)(hrow + k);
        v2f b = *(const v2f*)(wptr + k);
        // 8 args: (neg_a, A, neg_b, B, c_mod, C, reuse_a, reuse_b)
        c = __builtin_amdgcn_wmma_f32_16x16x4_f32(false, a, false, b,
                                                  (short)0, c, false, false);
    }

    if (m == 0) {                            // lanes 0 and 16 hold D column N=0
        float bias = bvec[level];
#pragma unroll
        for (int v = 0; v < 8; ++v) {
            int r = base + v + 8 * hi;
            if (r < n) {
                float dt = c[v];
                float gt = gate ? gate[r] : 1.0f;
                float x  = gt * dt + bias;
                scores[r] = 1.0f / (1.0f + expf(-x));
            }
        }
    }
}

// Exact descending-sort rank with lower-index tiebreak (matches jax.lax.top_k).
// rank(i) = #{ j : s_j > s_i  or (s_j == s_i and j < i) }; ranks are a
// permutation of 0..n-1, so each output slot < k is written exactly once.
__global__ __launch_bounds__(256)
void rank_select_kernel(const float* __restrict__ s, int n, int k,
                        int* __restrict__ idx, float* __restrict__ vals)
{
    __shared__ float sh[N_FULL];
    for (int j = threadIdx.x; j < n; j += 256) sh[j] = s[j];
    __syncthreads();
    int i = blockIdx.x * 256 + threadIdx.x;
    if (i >= n) return;
    float si = sh[i];
    int rank = 0;
    for (int j = 0; j < n; ++j) {            // LDS broadcast reads
        float sj = sh[j];
        rank += (sj > si) || (sj == si && j < i);
    }
    if (rank < k) { idx[rank] = i; vals[rank] = si; }
}

// sel_out[r] = chain-composed original-h row; gate_out[r] = product of gates.
__global__ __launch_bounds__(256)
void compose_kernel(const int* __restrict__ idx, const float* __restrict__ vals,
                    const int* __restrict__ sel_prev, const float* __restrict__ gate_prev,
                    int* __restrict__ sel_out, float* __restrict__ gate_out, int k)
{
    int r = blockIdx.x * 256 + threadIdx.x;
    if (r >= k) return;
    int p = idx[r];
    sel_out[r]  = sel_prev ? sel_prev[p] : p;
    gate_out[r] = (gate_prev ? gate_prev[p] : 1.0f) * vals[r];
}

// out[idx[r], :] += h[sel[r], :] * gate[r]   (one block per row, float4 lanes;
// rows within a level are distinct, levels run sequentially -> no atomics)
__global__ __launch_bounds__(128)
void scatter_kernel(const float* __restrict__ h, const int* __restrict__ idx,
                    const int* __restrict__ sel, const float* __restrict__ gate,
                    float* __restrict__ out, int k)
{
    int r = blockIdx.x;
    if (r >= k) return;
    int orow = idx[r];
    int srow = sel[r];
    float gt = gate[r];
    const float4* src = (const float4*)(h   + (size_t)srow * DIM);
    float4*       dst = (float4*)      (out + (size_t)orow * DIM);
    int d = threadIdx.x;                      // 0..127
    float4 v = src[d];
    float4 o = dst[d];
    o.x += v.x * gt; o.y += v.y * gt; o.z += v.z * gt; o.w += v.w * gt;
    dst[d] = o;
}

__global__ __launch_bounds__(256)
void zero_kernel(float4* __restrict__ out, int n4)
{
    int i = blockIdx.x * 256 + threadIdx.x;
    if (i < n4) out[i] = make_float4(0.f, 0.f, 0.f, 0.f);
}

extern "C" void kernel_launch(void* const* d_in, const int* in_sizes, int n_in,
                              void* d_out, int out_size, void* d_ws, size_t ws_size,
                              hipStream_t stream)
{
    (void)in_sizes; (void)n_in; (void)out_size; (void)ws_size;
    // inputs: g [N,N] (dead), h [N,D], W [3,D], b [3]
    const float* h = (const float*)d_in[1];
    const float* W = (const float*)d_in[2];
    const float* b = (const float*)d_in[3];
    float* out = (float*)d_out;

    // static level sizes, identical truncation semantics to Python int(k*n)
    const int n0 = N_FULL;
    int k0 = (int)(0.8 * (double)n0); if (k0 < 2) k0 = 2;   // 3276
    int k1 = (int)(0.6 * (double)k0); if (k1 < 2) k1 = 2;   // 1965
    int k2 = (int)(0.4 * (double)k1); if (k2 < 2) k2 = 2;   // 786

    // workspace carve (all 4-byte elements, ~100 KB total)
    float* fws    = (float*)d_ws;
    float* scores = fws;                 size_t off = N_FULL;
    int*   idx0   = (int*)(fws + off);   off += k0;
    float* vals0  = fws + off;           off += k0;
    int*   sel0   = (int*)(fws + off);   off += k0;
    float* gate0  = fws + off;           off += k0;
    int*   idx1   = (int*)(fws + off);   off += k1;
    float* vals1  = fws + off;           off += k1;
    int*   sel1   = (int*)(fws + off);   off += k1;
    float* gate1  = fws + off;           off += k1;
    int*   idx2   = (int*)(fws + off);   off += k2;
    float* vals2  = fws + off;           off += k2;
    int*   sel2   = (int*)(fws + off);   off += k2;
    float* gate2  = fws + off;           off += k2;

    const int n4 = N_FULL * DIM / 4;
    zero_kernel<<<(n4 + 255) / 256, 256, 0, stream>>>((float4*)out, n4);

    // ---- level 0 ----
    score_wmma_kernel<<<(n0 + 15) / 16, 32, 0, stream>>>(h, nullptr, nullptr,
                                                         W + 0 * DIM, b, 0, scores, n0);
    rank_select_kernel<<<(n0 + 255) / 256, 256, 0, stream>>>(scores, n0, k0, idx0, vals0);
    compose_kernel<<<(k0 + 255) / 256, 256, 0, stream>>>(idx0, vals0, nullptr, nullptr,
                                                         sel0, gate0, k0);
    scatter_kernel<<<k0, 128, 0, stream>>>(h, idx0, sel0, gate0, out, k0);

    // ---- level 1 ----
    score_wmma_kernel<<<(k0 + 15) / 16, 32, 0, stream>>>(h, sel0, gate0,
                                                         W + 1 * DIM, b, 1, scores, k0);
    rank_select_kernel<<<(k0 + 255) / 256, 256, 0, stream>>>(scores, k0, k1, idx1, vals1);
    compose_kernel<<<(k1 + 255) / 256, 256, 0, stream>>>(idx1, vals1, sel0, gate0,
                                                         sel1, gate1, k1);
    scatter_kernel<<<k1, 128, 0, stream>>>(h, idx1, sel1, gate1, out, k1);

    // ---- level 2 ----
    score_wmma_kernel<<<(k1 + 15) / 16, 32, 0, stream>>>(h, sel1, gate1,
                                                         W + 2 * DIM, b, 2, scores, k1);
    rank_select_kernel<<<(k1 + 255) / 256, 256, 0, stream>>>(scores, k1, k2, idx2, vals2);
    compose_kernel<<<(k2 + 255) / 256, 256, 0, stream>>>(idx2, vals2, sel1, gate1,
                                                         sel2, gate2, k2);
    scatter_kernel<<<k2, 128, 0, stream>>>(h, idx2, sel2, gate2, out, k2);
}